// KeyCompressor_33071248179491
// MI455X (gfx1250) — compile-verified
//
#include <hip/hip_runtime.h>
#include <hip/hip_bf16.h>

// Residual-VQ encode (CommVQ KeyCompressor) for MI455X / gfx1250.
// codes[r,h,n] = argmin_k || resid - cb[r,h,k] ||^2  ==  argmax_k (resid.cb_k - 0.5*||cb_k||^2)
// Heavy part: fp32 GEMM via V_WMMA_F32_16X16X4_F32 (native fp32 WMMA, wave32).
// Each wave owns a 32-row tile (two A panels + two interleaved accumulators) so every
// B-tile b64 load feeds two WMMAs and the two accumulate chains hide each other's latency.

#define B_  2
#define S_  2048
#define N_  (B_ * S_)   // 4096 rows (b*s)
#define H_  8
#define C_  128
#define R_  8
#define K_  4096

typedef __attribute__((ext_vector_type(2))) float v2f;
typedef __attribute__((ext_vector_type(8))) float v8f;

// ---------------- preprocess: interleave permutation + L2 normalize ----------------
// x[n, h*128 + q*64 + p]  ->  resid[h][n][2p+q] / ||x[n,:]||
__global__ __launch_bounds__(256)
void vq_prep(const float* __restrict__ x, float* __restrict__ resid,
             float* __restrict__ prescale) {
  const int n = blockIdx.x;        // row in [0, N)
  const int t = threadIdx.x;       // 0..255
  __shared__ float red[256];
  const float* xr = x + (size_t)n * (H_ * C_);

  float vals[4];
  float ss = 0.f;
#pragma unroll
  for (int i = 0; i < 4; ++i) {
    float v = xr[t + 256 * i];
    vals[i] = v;
    ss += v * v;
  }
  red[t] = ss;
  __syncthreads();
  for (int off = 128; off > 0; off >>= 1) {
    if (t < off) red[t] += red[t + off];
    __syncthreads();
  }
  const float nrm = __builtin_sqrtf(red[0]);
  if (t == 0) prescale[n] = nrm;
  const float inv = 1.0f / nrm;
#pragma unroll
  for (int i = 0; i < 4; ++i) {
    const int u   = t + 256 * i;      // position in original 1024-vector
    const int h   = u >> 7;
    const int rem = u & 127;
    const int q   = rem >> 6;         // real/imag block
    const int p   = rem & 63;         // position within block
    const int j   = 2 * p + q;        // interleaved position
    resid[((size_t)(h * N_) + n) * C_ + j] = vals[i] * inv;
  }
}

// ---------------- codebook row squared norms: cnorm[r*H*K + ...] ----------------
__global__ __launch_bounds__(256)
void vq_cnorm(const float* __restrict__ cb, float* __restrict__ cnorm) {
  const int wave = (int)((blockIdx.x * blockDim.x + threadIdx.x) >> 5); // one row per wave
  const int lane = threadIdx.x & 31;
  const float* row = cb + (size_t)wave * C_;
  float ss = 0.f;
#pragma unroll
  for (int i = 0; i < 4; ++i) {
    float v = row[lane + 32 * i];
    ss += v * v;
  }
#pragma unroll
  for (int m = 16; m >= 1; m >>= 1) ss += __shfl_xor(ss, m, 32);
  if (lane == 0) cnorm[wave] = ss;
}

// ---------------- one RVQ stage: distance GEMM + argmin + residual update ----------------
__global__ __launch_bounds__(256)
void vq_stage(const float* __restrict__ cb,     // [R,H,K,C]
              const float* __restrict__ cnorm,  // [R*H*K]
              float* __restrict__ resid,        // [H,N,C]
              int* __restrict__ codes,          // [R,H,N]
              int r) {
  const int lane    = threadIdx.x & 31;
  const int wave    = threadIdx.x >> 5;
  const int tile    = blockIdx.x * 8 + wave;    // 0..1023  (H * N/32)
  const int h       = tile >> 7;                // tile / 128
  const int rowBase = (tile & 127) << 5;        // 32 rows per tile

  const float* cbr    = cb    + ((size_t)(r * H_ + h) * K_) * C_;
  const float* cnr    = cnorm + (size_t)(r * H_ + h) * K_;
  float*       residH = resid + (size_t)h * N_ * C_;

  const int halfSel = lane >> 4;   // 0: lanes 0-15, 1: lanes 16-31
  const int lane15  = lane & 15;
  const int koff    = halfSel * 2; // fp32 WMMA A/B: low half holds k{0,1}, high half k{2,3}

  // Two A panels: rows [rowBase, rowBase+16) and [rowBase+16, rowBase+32).
  v2f a0[32], a1[32];
  {
    const float* arow0 = residH + (size_t)(rowBase + lane15) * C_ + koff;
    const float* arow1 = arow0 + 16 * C_;
#pragma unroll
    for (int j = 0; j < 32; ++j) {
      a0[j] = *(const v2f*)(arow0 + 4 * j);
      a1[j] = *(const v2f*)(arow1 + 4 * j);
    }
  }

  float bestv0[8], bestv1[8];
  int   besti0[8], besti1[8];
#pragma unroll
  for (int i = 0; i < 8; ++i) {
    bestv0[i] = -__builtin_inff(); besti0[i] = 0;
    bestv1[i] = -__builtin_inff(); besti1[i] = 0;
  }

  for (int kb = 0; kb < K_; kb += 16) {
    const int    code = kb + lane15;                     // B column owned by this lane
    const float* brow = cbr + (size_t)code * C_ + koff;  // B(k,n) = cb[code=n][dim=k]
    const float  cn   = cnr[code];
    v8f acc0 = {}, acc1 = {};
#pragma unroll
    for (int j = 0; j < 32; ++j) {
      v2f b = *(const v2f*)(brow + 4 * j);
      // One B load feeds two independent accumulate chains (latency hiding).
      acc0 = __builtin_amdgcn_wmma_f32_16x16x4_f32(false, a0[j], false, b,
                                                   (short)0, acc0, false, false);
      acc1 = __builtin_amdgcn_wmma_f32_16x16x4_f32(false, a1[j], false, b,
                                                   (short)0, acc1, false, false);
    }
    // score = dot - 0.5*||cb||^2 ; argmax == argmin of squared distance.
    // Strict '>' keeps the earliest (lowest) code on in-lane ties, matching jnp.argmin.
    const float hc = 0.5f * cn;
#pragma unroll
    for (int i = 0; i < 8; ++i) {
      float m0 = acc0[i] - hc;
      float m1 = acc1[i] - hc;
      if (m0 > bestv0[i]) { bestv0[i] = m0; besti0[i] = code; }
      if (m1 > bestv1[i]) { bestv1[i] = m1; besti1[i] = code; }
    }
  }

  // C layout: vgpr i of acc0 holds row (rowBase + halfSel*8 + i), acc1 the same +16;
  // columns are striped over 16 lanes. Reduce (val,idx) argmax across each 16-lane half
  // (xor masks 1..8 never cross the half boundary).
#pragma unroll
  for (int i = 0; i < 8; ++i) {
    float v0 = bestv0[i]; int ix0 = besti0[i];
    float v1 = bestv1[i]; int ix1 = besti1[i];
#pragma unroll
    for (int m = 1; m < 16; m <<= 1) {
      float ov0 = __shfl_xor(v0, m, 32);
      int   oi0 = __shfl_xor(ix0, m, 32);
      float ov1 = __shfl_xor(v1, m, 32);
      int   oi1 = __shfl_xor(ix1, m, 32);
      if (ov0 > v0 || (ov0 == v0 && oi0 < ix0)) { v0 = ov0; ix0 = oi0; }
      if (ov1 > v1 || (ov1 == v1 && oi1 < ix1)) { v1 = ov1; ix1 = oi1; }
    }
    bestv0[i] = v0; besti0[i] = ix0;
    bestv1[i] = v1; besti1[i] = ix1;
  }

  if (lane15 == 0) {
    int* crow = codes + (size_t)(r * H_ + h) * N_ + rowBase + halfSel * 8;
#pragma unroll
    for (int i = 0; i < 8; ++i) {
      crow[i]      = besti0[i];
      crow[i + 16] = besti1[i];
    }
  }

  // Residual update for next stage: resid[row] -= cb[winner(row)]; wave owns its 32 rows.
  for (int M = 0; M < 32; ++M) {
    const int srcLane = ((M >> 3) & 1) << 4;   // half that holds this row's result
    const int ii      = M & 7;
    const int widx    = (M < 16) ? __shfl(besti0[ii], srcLane, 32)
                                 : __shfl(besti1[ii], srcLane, 32);
    const float* crow = cbr + (size_t)widx * C_;
    float*       rrow = residH + (size_t)(rowBase + M) * C_;
#pragma unroll
    for (int d = 0; d < 4; ++d)
      rrow[lane + 32 * d] -= crow[lane + 32 * d];
  }
}

extern "C" void kernel_launch(void* const* d_in, const int* in_sizes, int n_in,
                              void* d_out, int out_size, void* d_ws, size_t ws_size,
                              hipStream_t stream) {
  const float* x  = (const float*)d_in[0];   // [B,S,H*C]
  const float* cb = (const float*)d_in[1];   // [R,H,K,C]

  int*   codes    = (int*)d_out;                          // R*H*N int32
  float* prescale = (float*)d_out + (size_t)R_ * H_ * N_; // N float32

  float* resid = (float*)d_ws;                         // H*N*C floats (16 MB)
  float* cnorm = resid + (size_t)H_ * N_ * C_;         // R*H*K floats (1 MB)

  vq_prep <<<N_, 256, 0, stream>>>(x, resid, prescale);
  vq_cnorm<<<(R_ * H_ * K_) / 8, 256, 0, stream>>>(cb, cnorm);

  // Stages are serially dependent (residual update); same-stream launches order them.
  for (int r = 0; r < R_; ++r)
    vq_stage<<<(H_ * (N_ / 32)) / 8, 256, 0, stream>>>(cb, cnorm, resid, codes, r);
}